// GCNModel_2645699854673
// MI455X (gfx1250) — compile-verified
//
#include <hip/hip_runtime.h>
#include <hip/hip_bf16.h>

// ---------------------------------------------------------------------------
// GCN (3-layer GraphConv, norm='both') for MI455X / gfx1250.
// Dominant cost: edge scatter-add (memory/atomic bound). Dense layer GEMMs use
// full-precision V_WMMA_F32_16X16X4_F32 (fp32 in/out, matches reference math).
// ---------------------------------------------------------------------------

typedef __attribute__((ext_vector_type(2))) float v2f;
typedef __attribute__((ext_vector_type(8))) float v8f;

#define HIDDEN 64

// ---- zero a float4-aligned region -----------------------------------------
__global__ void zero4_kernel(float4* __restrict__ p, int n4) {
    int t = blockIdx.x * blockDim.x + threadIdx.x;
    if (t < n4) p[t] = make_float4(0.f, 0.f, 0.f, 0.f);
}

// ---- degree accumulation: one thread per edge ------------------------------
__global__ void degree_kernel(const int* __restrict__ src, const int* __restrict__ dst,
                              float* __restrict__ out_deg, float* __restrict__ in_deg,
                              int n_edges) {
    int e = blockIdx.x * blockDim.x + threadIdx.x;
    if (e >= n_edges) return;
    atomicAdd(out_deg + src[e], 1.0f);
    atomicAdd(in_deg + dst[e], 1.0f);
}

// ---- in-place deg -> rsqrt(max(deg,1)) over both arrays (2N contiguous) ----
__global__ void rsqrt_kernel(float* __restrict__ v, int n) {
    int t = blockIdx.x * blockDim.x + threadIdx.x;
    if (t < n) v[t] = rsqrtf(fmaxf(v[t], 1.0f));
}

// ---- embedding gather: h[i][:] = emb_table[feats[i]][:] (16 thr / node) ----
__global__ void embed_kernel(const int* __restrict__ feats,
                             const float* __restrict__ emb,
                             float* __restrict__ h, int n_nodes) {
    int t = blockIdx.x * blockDim.x + threadIdx.x;
    int i = t >> 4;
    if (i >= n_nodes) return;
    int c = (t & 15) * 4;
    const float4 v = *(const float4*)(emb + (size_t)feats[i] * HIDDEN + c);
    *(float4*)(h + (size_t)i * HIDDEN + c) = v;
}

// ---- SpMM scatter: agg[dst] += h[src] * norm_src[src] (16 thr / edge) ------
__global__ void scatter_kernel(const int* __restrict__ src, const int* __restrict__ dst,
                               const float* __restrict__ h,
                               const float* __restrict__ norm_src,
                               float* __restrict__ agg, int n_edges) {
    int t = blockIdx.x * blockDim.x + threadIdx.x;
    int e = t >> 4;
    if (e >= n_edges) return;
    int c = (t & 15) * 4;
    int s = src[e];
    int d = dst[e];
    float ns = norm_src[s];
    const float4 v = *(const float4*)(h + (size_t)s * HIDDEN + c);
    float* out = agg + (size_t)d * HIDDEN + c;
    atomicAdd(out + 0, v.x * ns);
    atomicAdd(out + 1, v.y * ns);
    atomicAdd(out + 2, v.z * ns);
    atomicAdd(out + 3, v.w * ns);
}

// ---- layer GEMM via fp32 WMMA: h = (agg * norm_dst) @ W + b ----------------
// One wave -> one 16-row x 64-col output tile.
// A frag (16x4 f32): lanes 0-15 hold row m=l16 K={0,1}; lanes 16-31 K={2,3}.
// B frag (4x16 f32): lane = column n; half selects K pair. C/D: v[r] =
// D[r + 8*half][l16].
__global__ void gemm_wmma_kernel(const float* __restrict__ agg,
                                 const float* __restrict__ norm_dst,
                                 const float* __restrict__ W,     // [64][64] k-major
                                 const float* __restrict__ bias,  // [64]
                                 float* __restrict__ h_out,
                                 int n_nodes) {
    const int lane = threadIdx.x & 31;
    const int wave = threadIdx.x >> 5;
    const int wavesPerBlock = blockDim.x >> 5;
    const int tile = blockIdx.x * wavesPerBlock + wave;
    const int row0 = tile * 16;
    if (row0 >= n_nodes) return;  // uniform across the wave

    const int half = lane >> 4;  // 0: K pair {0,1}; 1: K pair {2,3}
    const int l16  = lane & 15;

    int mrow = row0 + l16;
    if (mrow >= n_nodes) mrow = n_nodes - 1;  // safety (N % 16 == 0 normally)
    const float scale = norm_dst[mrow];
    const float* __restrict__ arow = agg + (size_t)mrow * HIDDEN;

    // Preload all A fragments (16 k-steps), folding in the norm_dst scaling.
    v2f afrag[16];
#pragma unroll
    for (int s = 0; s < 16; ++s) {
        const int kb = 4 * s + 2 * half;
        afrag[s].x = arow[kb] * scale;
        afrag[s].y = arow[kb + 1] * scale;
    }

    v8f acc[4];
#pragma unroll
    for (int nt = 0; nt < 4; ++nt) acc[nt] = (v8f){0.f, 0.f, 0.f, 0.f, 0.f, 0.f, 0.f, 0.f};

#pragma unroll
    for (int nt = 0; nt < 4; ++nt) {
        const int n = nt * 16 + l16;
#pragma unroll
        for (int s = 0; s < 16; ++s) {
            const int kb = 4 * s + 2 * half;
            v2f b;
            b.x = W[(size_t)kb * HIDDEN + n];
            b.y = W[(size_t)(kb + 1) * HIDDEN + n];
            acc[nt] = __builtin_amdgcn_wmma_f32_16x16x4_f32(
                false, afrag[s], false, b, (short)0, acc[nt], false, false);
        }
    }

    // Bias add + store. Lane holds column n = nt*16 + l16; rows r + 8*half.
#pragma unroll
    for (int nt = 0; nt < 4; ++nt) {
        const int n = nt * 16 + l16;
        const float bn = bias[n];
#pragma unroll
        for (int r = 0; r < 8; ++r) {
            const int m = row0 + r + 8 * half;
            if (m < n_nodes) h_out[(size_t)m * HIDDEN + n] = acc[nt][r] + bn;
        }
    }
}

// ---- sum pooling: pooled[graph_ids[i]] += h[i] (16 thr / node) -------------
__global__ void pool_kernel(const int* __restrict__ gid, const float* __restrict__ h,
                            float* __restrict__ pooled, int n_nodes) {
    int t = blockIdx.x * blockDim.x + threadIdx.x;
    int i = t >> 4;
    if (i >= n_nodes) return;
    int c = (t & 15) * 4;
    int g = gid[i];
    const float4 v = *(const float4*)(h + (size_t)i * HIDDEN + c);
    float* out = pooled + (size_t)g * HIDDEN + c;
    atomicAdd(out + 0, v.x);
    atomicAdd(out + 1, v.y);
    atomicAdd(out + 2, v.z);
    atomicAdd(out + 3, v.w);
}

// ---- regression head: out[g] = pooled[g] . W_reg ---------------------------
__global__ void head_kernel(const float* __restrict__ pooled,
                            const float* __restrict__ Wreg,
                            float* __restrict__ out, int n_graphs) {
    int g = blockIdx.x * blockDim.x + threadIdx.x;
    if (g >= n_graphs) return;
    const float* p = pooled + (size_t)g * HIDDEN;
    float acc = 0.f;
#pragma unroll
    for (int k = 0; k < HIDDEN; k += 4) {
        const float4 a = *(const float4*)(p + k);
        const float4 w = *(const float4*)(Wreg + k);
        acc = fmaf(a.x, w.x, acc);
        acc = fmaf(a.y, w.y, acc);
        acc = fmaf(a.z, w.z, acc);
        acc = fmaf(a.w, w.w, acc);
    }
    out[g] = acc;
}

static inline int cdiv(long long a, long long b) { return (int)((a + b - 1) / b); }

extern "C" void kernel_launch(void* const* d_in, const int* in_sizes, int n_in,
                              void* d_out, int out_size, void* d_ws, size_t ws_size,
                              hipStream_t stream) {
    const int*   feats = (const int*)d_in[0];
    const int*   src   = (const int*)d_in[1];
    const int*   dst   = (const int*)d_in[2];
    const int*   gid   = (const int*)d_in[3];
    /* d_in[4] = n_graphs scalar on device; OUT_DIM==1 so out_size == n_graphs */
    const float* emb   = (const float*)d_in[5];
    const float* Ws    = (const float*)d_in[6];  // [3][64][64]
    const float* bs    = (const float*)d_in[7];  // [3][64]
    const float* Wreg  = (const float*)d_in[8];  // [64]
    float*       out   = (float*)d_out;

    const int N = in_sizes[0];
    const int E = in_sizes[1];
    const int LAYERS = in_sizes[7] / HIDDEN;     // bs flat size / HIDDEN
    const int n_graphs = out_size;               // OUT_DIM == 1

    // Workspace layout (all offsets 16B-aligned; N*4 bytes is multiple of 16).
    float* ws       = (float*)d_ws;
    float* norm_src = ws;                         // [N]
    float* norm_dst = ws + (size_t)N;             // [N]
    float* h        = ws + 2 * (size_t)N;         // [N][64]
    float* agg      = h + (size_t)N * HIDDEN;     // [N][64]
    float* pooled   = agg + (size_t)N * HIDDEN;   // [n_graphs][64]
    // total: (2 + 128) * N + 64 * n_graphs floats  (~52 MB for the reference)

    const int B = 256;

    // 1) degrees -> rsqrt norms
    zero4_kernel<<<cdiv(2LL * N / 4, B), B, 0, stream>>>((float4*)norm_src, 2 * N / 4);
    degree_kernel<<<cdiv(E, B), B, 0, stream>>>(src, dst, norm_src, norm_dst, E);
    rsqrt_kernel<<<cdiv(2LL * N, B), B, 0, stream>>>(norm_src, 2 * N);

    // 2) embedding gather
    embed_kernel<<<cdiv((long long)N * 16, B), B, 0, stream>>>(feats, emb, h, N);

    // 3) graph-conv layers
    const int tiles = cdiv(N, 16);
    const int gemm_blocks = cdiv(tiles, B / 32);
    for (int l = 0; l < LAYERS; ++l) {
        zero4_kernel<<<cdiv((long long)N * HIDDEN / 4, B), B, 0, stream>>>(
            (float4*)agg, N * HIDDEN / 4);
        scatter_kernel<<<cdiv((long long)E * 16, B), B, 0, stream>>>(
            src, dst, h, norm_src, agg, E);
        gemm_wmma_kernel<<<gemm_blocks, B, 0, stream>>>(
            agg, norm_dst, Ws + (size_t)l * HIDDEN * HIDDEN, bs + (size_t)l * HIDDEN,
            h, N);
    }

    // 4) sum pooling + regression head
    zero4_kernel<<<cdiv((long long)n_graphs * HIDDEN / 4, B), B, 0, stream>>>(
        (float4*)pooled, n_graphs * HIDDEN / 4);
    pool_kernel<<<cdiv((long long)N * 16, B), B, 0, stream>>>(gid, h, pooled, N);
    head_kernel<<<cdiv(n_graphs, B), B, 0, stream>>>(pooled, Wreg, out, n_graphs);
}